// TGN_36026185678963
// MI455X (gfx1250) — compile-verified
//
#include <hip/hip_runtime.h>
#include <math.h>

// ---------------------------------------------------------------------------
// TGN-style GAT forward for MI455X (gfx1250, wave32).
//   - GEMMs use V_WMMA_F32_16X16X4_F32 (exact fp32 matrix math on the XDL pipe)
//   - mem==0  =>  node-proj + concat + W1 folded into ONE GEMM: x @ (node_w@w1_top)
//   - edge softmax/aggregation via hardware float atomics (add) and
//     order-preserving uint encoding for float atomicMax
// ---------------------------------------------------------------------------

typedef __attribute__((ext_vector_type(2))) float v2f;
typedef __attribute__((ext_vector_type(8))) float v8f;

#define HID 128
#define GK 128          // GEMM K
#define GN 256          // GEMM N (= HEADS*HID)
#define NUM_GRAPHS 64

// ---- order-preserving float <-> uint mapping (for atomicMax on floats) -----
__device__ __forceinline__ unsigned encf(float f) {
    unsigned u = __float_as_uint(f);
    return (u >> 31) ? ~u : (u | 0x80000000u);
}
__device__ __forceinline__ float decf(unsigned v) {
    return (v >> 31) ? __uint_as_float(v & 0x7FFFFFFFu) : __uint_as_float(~v);
}
#define ENC_NEG_INF 0x007FFFFFu   // encf(-inf)

__device__ __forceinline__ float fatomic_add(float* p, float v) {
    return unsafeAtomicAdd(p, v);   // -> global_atomic_add_f32
}

// ---------------------------------------------------------------------------
// init helpers
// ---------------------------------------------------------------------------
__global__ void k_init_f32(float* p, int n, float v) {
    int i = blockIdx.x * blockDim.x + threadIdx.x;
    if (i < n) p[i] = v;
}
__global__ void k_init_u32(unsigned* p, int n, unsigned v) {
    int i = blockIdx.x * blockDim.x + threadIdx.x;
    if (i < n) p[i] = v;
}

// ---------------------------------------------------------------------------
// Precompute Wc = node_w @ w1_top  (128x256), bc = node_b @ w1_top (256)
// (w1 is [228,256]; only rows 0..127 matter since mem==0)
// ---------------------------------------------------------------------------
__global__ void k_prep_wc(const float* __restrict__ nodew,
                          const float* __restrict__ w1,
                          float* __restrict__ Wc) {
    int c = threadIdx.x;          // 0..255
    int r = blockIdx.x;           // 0..127
    float acc = 0.f;
    for (int k = 0; k < 128; ++k) acc += nodew[r * 128 + k] * w1[k * 256 + c];
    Wc[r * 256 + c] = acc;
}
__global__ void k_prep_bc(const float* __restrict__ nodeb,
                          const float* __restrict__ w1,
                          float* __restrict__ bc) {
    int c = threadIdx.x;          // 0..255
    float acc = 0.f;
    for (int k = 0; k < 128; ++k) acc += nodeb[k] * w1[k * 256 + c];
    bc[c] = acc;
}

// ---------------------------------------------------------------------------
// GEMM: C[M,256] = A[M,128] @ W[128,256] (+ bias[256])     M % 16 == 0
// 256 threads = 8 waves; wave owns a 16-row stripe, A tile held in registers,
// W column tiles (128x16 = 8KB) staged in LDS.  v_wmma_f32_16x16x4_f32.
// ---------------------------------------------------------------------------
__global__ __launch_bounds__(256) void k_gemm(const float* __restrict__ A,
                                              const float* __restrict__ W,
                                              const float* __restrict__ bias,
                                              float* __restrict__ C, int M) {
    __shared__ float sW[GK * 16];                 // one 128x16 column tile
    const int t    = threadIdx.x;
    const int wave = t >> 5;
    const int lane = t & 31;

    const int row0   = blockIdx.x * 128 + wave * 16;
    const bool active = (row0 < M);
    const int rowc   = active ? row0 : (M - 16);  // clamp (M%16==0)

    const int koff = (lane >> 4) * 2;             // lanes 16-31 hold K+2,K+3
    const int mrow = lane & 15;

    // A fragments for all 32 K-steps (16x4 f32 layout from the ISA)
    const float2* ap2 = (const float2*)(A + (size_t)(rowc + mrow) * GK + koff);
    v2f afr[32];
#pragma unroll
    for (int ks = 0; ks < 32; ++ks) {
        float2 a2 = ap2[ks * 2];                  // stride 4 floats
        afr[ks].x = a2.x;
        afr[ks].y = a2.y;
    }

    const int r     = t >> 1;                     // W staging: row 0..127
    const int cpart = (t & 1) * 8;                // 0 or 8

    for (int ct = 0; ct < 16; ++ct) {
        __syncthreads();
        const float4* wsrc = (const float4*)(W + (size_t)r * GN + ct * 16 + cpart);
        float4 w0 = wsrc[0];
        float4 w1v = wsrc[1];
        ((float4*)(sW + r * 16 + cpart))[0] = w0;
        ((float4*)(sW + r * 16 + cpart))[1] = w1v;
        __syncthreads();

        v8f acc = {};
#pragma unroll
        for (int ks = 0; ks < 32; ++ks) {
            v2f b;
            b.x = sW[(ks * 4 + koff + 0) * 16 + mrow];
            b.y = sW[(ks * 4 + koff + 1) * 16 + mrow];
            acc = __builtin_amdgcn_wmma_f32_16x16x4_f32(
                false, afr[ks], false, b, (short)0, acc, false, false);
        }

        if (active) {
            const int col   = ct * 16 + mrow;
            const float bv  = bias ? bias[col] : 0.f;
            const int rbase = row0 + (lane >> 4) * 8;
#pragma unroll
            for (int i = 0; i < 8; ++i)
                C[(size_t)(rbase + i) * GN + col] = acc[i] + bv;
        }
    }
}

// ---------------------------------------------------------------------------
// Per-node attention scalars: als[n,h] = x1[n,h,:] . a_src[h], same for dst.
// One wave per node; lanes 0-15 = head 0, lanes 16-31 = head 1.
// ---------------------------------------------------------------------------
__global__ void k_attn_scalars(const float* __restrict__ x,
                               const float* __restrict__ asrc,
                               const float* __restrict__ adst,
                               float* __restrict__ als, float* __restrict__ ald,
                               int Nn) {
    int gt = blockIdx.x * blockDim.x + threadIdx.x;
    int wave = gt >> 5, lane = gt & 31;
    if (wave >= Nn) return;
    const float4* xr = (const float4*)(x + (size_t)wave * 256 + lane * 8);
    const float4* ar = (const float4*)(asrc + lane * 8);
    const float4* dr = (const float4*)(adst + lane * 8);
    float4 x0 = xr[0], x1 = xr[1];
    float4 a0 = ar[0], a1 = ar[1];
    float4 d0 = dr[0], d1 = dr[1];
    float ss = x0.x * a0.x + x0.y * a0.y + x0.z * a0.z + x0.w * a0.w
             + x1.x * a1.x + x1.y * a1.y + x1.z * a1.z + x1.w * a1.w;
    float dd = x0.x * d0.x + x0.y * d0.y + x0.z * d0.z + x0.w * d0.w
             + x1.x * d1.x + x1.y * d1.y + x1.z * d1.z + x1.w * d1.w;
#pragma unroll
    for (int o = 8; o >= 1; o >>= 1) {            // reduce within 16-lane half
        ss += __shfl_xor(ss, o, 32);
        dd += __shfl_xor(dd, o, 32);
    }
    if ((lane & 15) == 0) {
        int h = lane >> 4;
        als[wave * 2 + h] = ss;
        ald[wave * 2 + h] = dd;
    }
}

__device__ __forceinline__ void edge_sd(const int* ei, int E, int e, int& s, int& d) {
    if (e < E) { s = ei[e]; d = ei[E + e]; }
    else       { s = d = e - E; }                 // self loop
}
__device__ __forceinline__ float lrelu(float t) { return t > 0.f ? t : 0.2f * t; }

// pass 1: segment max of attention logits (ordered-uint atomicMax)
__global__ void k_edge_max(const int* __restrict__ ei, int E, int ET,
                           const float* __restrict__ als,
                           const float* __restrict__ ald,
                           unsigned* __restrict__ menc) {
    int e = blockIdx.x * blockDim.x + threadIdx.x;
    if (e >= ET) return;
    int s, d; edge_sd(ei, E, e, s, d);
#pragma unroll
    for (int h = 0; h < 2; ++h) {
        float t = lrelu(als[s * 2 + h] + ald[d * 2 + h]);
        atomicMax(&menc[d * 2 + h], encf(t));
    }
}

// pass 2: denom[d,h] = sum exp(e - m[d,h])
__global__ void k_edge_expsum(const int* __restrict__ ei, int E, int ET,
                              const float* __restrict__ als,
                              const float* __restrict__ ald,
                              const unsigned* __restrict__ menc,
                              float* __restrict__ denom) {
    int e = blockIdx.x * blockDim.x + threadIdx.x;
    if (e >= ET) return;
    int s, d; edge_sd(ei, E, e, s, d);
#pragma unroll
    for (int h = 0; h < 2; ++h) {
        float t = lrelu(als[s * 2 + h] + ald[d * 2 + h]);
        float m = decf(menc[d * 2 + h]);
        fatomic_add(&denom[d * 2 + h], __expf(t - m));
    }
}

// pass 3: agg[d,h,:] += alpha * x[s,h,:]   (one wave per edge, 8 floats/lane)
__global__ void k_edge_agg(const int* __restrict__ ei, int E, int ET,
                           const float* __restrict__ x,
                           const float* __restrict__ als,
                           const float* __restrict__ ald,
                           const unsigned* __restrict__ menc,
                           const float* __restrict__ denom,
                           float* __restrict__ agg) {
    int gt = blockIdx.x * blockDim.x + threadIdx.x;
    int wave = gt >> 5, lane = gt & 31;
    if (wave >= ET) return;
    int s, d; edge_sd(ei, E, wave, s, d);
    int h = lane >> 4;
    float t = lrelu(als[s * 2 + h] + ald[d * 2 + h]);
    float m = decf(menc[d * 2 + h]);
    float alpha = __expf(t - m) / (denom[d * 2 + h] + 1e-16f);
    const float4* xs = (const float4*)(x + (size_t)s * 256 + lane * 8);
    float4 v0 = xs[0], v1 = xs[1];
    float* ag = agg + (size_t)d * 256 + lane * 8;
    fatomic_add(ag + 0, v0.x * alpha);
    fatomic_add(ag + 1, v0.y * alpha);
    fatomic_add(ag + 2, v0.z * alpha);
    fatomic_add(ag + 3, v0.w * alpha);
    fatomic_add(ag + 4, v1.x * alpha);
    fatomic_add(ag + 5, v1.y * alpha);
    fatomic_add(ag + 6, v1.z * alpha);
    fatomic_add(ag + 7, v1.w * alpha);
}

// h[n,c] = relu( 0.5*(agg[n,0,c]+agg[n,1,c]) + b[c] )
__global__ void k_finalize(const float* __restrict__ agg,
                           const float* __restrict__ b,
                           float* __restrict__ h, int Nn) {
    int i = blockIdx.x * blockDim.x + threadIdx.x;
    if (i >= Nn * HID) return;
    int n = i >> 7, c = i & 127;
    float v = 0.5f * (agg[(size_t)n * 256 + c] + agg[(size_t)n * 256 + 128 + c]) + b[c];
    h[i] = fmaxf(v, 0.f);
}

// graph pooling: mean-sum + max (+ node counts)
__global__ void k_pool(const float* __restrict__ h, const int* __restrict__ batch,
                       float* __restrict__ meanacc, unsigned* __restrict__ maxenc,
                       unsigned* __restrict__ cnt, int Nn) {
    int i = blockIdx.x * blockDim.x + threadIdx.x;
    if (i >= Nn * HID) return;
    int n = i >> 7, c = i & 127;
    int g = batch[n];
    float v = h[i];
    fatomic_add(&meanacc[g * HID + c], v);
    atomicMax(&maxenc[g * HID + c], encf(v));
    if (c == 0) atomicAdd(&cnt[g], 1u);
}

// classifier: out[g,:] = relu([mean,max] @ wc1 + bc1) @ wc2 + bc2
__global__ __launch_bounds__(256) void k_classifier(
        const float* __restrict__ meanacc, const unsigned* __restrict__ maxenc,
        const unsigned* __restrict__ cnt,
        const float* __restrict__ wc1, const float* __restrict__ bc1,
        const float* __restrict__ wc2, const float* __restrict__ bc2,
        float* __restrict__ out) {
    __shared__ float hg[256];
    __shared__ float z1[128];
    int g = blockIdx.x, t = threadIdx.x;
    if (t < 128) {
        float c = fmaxf((float)cnt[g], 1.0f);
        hg[t] = meanacc[g * HID + t] / c;
    } else {
        unsigned v = maxenc[g * HID + (t - 128)];
        hg[t] = (v == ENC_NEG_INF) ? 0.f : decf(v);   // where(isfinite(max),max,0)
    }
    __syncthreads();
    if (t < 128) {
        float acc = bc1[t];
        for (int j = 0; j < 256; ++j) acc += hg[j] * wc1[j * 128 + t];
        z1[t] = fmaxf(acc, 0.f);
    }
    __syncthreads();
    if (t < 2) {
        float acc = bc2[t];
        for (int j = 0; j < 128; ++j) acc += z1[j] * wc2[j * 2 + t];
        out[g * 2 + t] = acc;
    }
}

// ---------------------------------------------------------------------------
// launch
// ---------------------------------------------------------------------------
static inline int cdiv(int a, int b) { return (a + b - 1) / b; }

extern "C" void kernel_launch(void* const* d_in, const int* in_sizes, int n_in,
                              void* d_out, int out_size, void* d_ws, size_t ws_size,
                              hipStream_t stream) {
    const float* x      = (const float*)d_in[0];
    const int*   ei     = (const int*)d_in[1];
    const int*   batch  = (const int*)d_in[2];
    const float* node_w = (const float*)d_in[3];
    const float* node_b = (const float*)d_in[4];
    const float* w1     = (const float*)d_in[5];
    const float* a1s    = (const float*)d_in[6];
    const float* a1d    = (const float*)d_in[7];
    const float* b1     = (const float*)d_in[8];
    const float* w2     = (const float*)d_in[9];
    const float* a2s    = (const float*)d_in[10];
    const float* a2d    = (const float*)d_in[11];
    const float* b2     = (const float*)d_in[12];
    const float* wc1    = (const float*)d_in[13];
    const float* bc1    = (const float*)d_in[14];
    const float* wc2    = (const float*)d_in[15];
    const float* bc2    = (const float*)d_in[16];
    float* out = (float*)d_out;

    const int N  = in_sizes[0] / 128;     // 50000
    const int E  = in_sizes[1] / 2;       // 800000
    const int ET = E + N;                 // edges + self loops

    // ---- workspace carve (~130 MB) ----
    size_t off = 0;
    auto carve = [&](size_t bytes) {
        void* p = (char*)d_ws + off;
        off += (bytes + 255) & ~(size_t)255;
        return p;
    };
    float*    xF      = (float*)carve((size_t)N * 256 * 4);  // x1 / x2
    float*    agg     = (float*)carve((size_t)N * 256 * 4);
    float*    hbuf    = (float*)carve((size_t)N * 128 * 4);  // h2 / h3
    float*    Wc      = (float*)carve(128 * 256 * 4);
    float*    bc      = (float*)carve(256 * 4);
    float*    als     = (float*)carve((size_t)N * 2 * 4);
    float*    ald     = (float*)carve((size_t)N * 2 * 4);
    unsigned* menc    = (unsigned*)carve((size_t)N * 2 * 4);
    float*    denom   = (float*)carve((size_t)N * 2 * 4);
    float*    meanacc = (float*)carve(NUM_GRAPHS * HID * 4);
    unsigned* maxenc  = (unsigned*)carve(NUM_GRAPHS * HID * 4);
    unsigned* cnt     = (unsigned*)carve(NUM_GRAPHS * 4);
    (void)ws_size; (void)n_in; (void)out_size;

    const int BT = 256;
    const int gemmBlocks = cdiv(N, 128);

    // ---- front end: fold node_proj + mem-concat + W1 into one GEMM ----
    k_prep_wc<<<128, 256, 0, stream>>>(node_w, w1, Wc);
    k_prep_bc<<<1, 256, 0, stream>>>(node_b, w1, bc);
    k_gemm<<<gemmBlocks, BT, 0, stream>>>(x, Wc, bc, xF, N);

    // ---- GAT layer (runs twice) ----
    auto gat_attention = [&](const float* asrc, const float* adst, const float* bvec) {
        k_attn_scalars<<<cdiv(N * 32, BT), BT, 0, stream>>>(xF, asrc, adst, als, ald, N);
        k_init_u32<<<cdiv(N * 2, BT), BT, 0, stream>>>(menc, N * 2, ENC_NEG_INF);
        k_init_f32<<<cdiv(N * 2, BT), BT, 0, stream>>>(denom, N * 2, 0.f);
        k_init_f32<<<cdiv(N * 256, BT), BT, 0, stream>>>(agg, N * 256, 0.f);
        k_edge_max<<<cdiv(ET, BT), BT, 0, stream>>>(ei, E, ET, als, ald, menc);
        k_edge_expsum<<<cdiv(ET, BT), BT, 0, stream>>>(ei, E, ET, als, ald, menc, denom);
        k_edge_agg<<<cdiv(ET * 32, BT), BT, 0, stream>>>(ei, E, ET, xF, als, ald, menc, denom, agg);
        k_finalize<<<cdiv(N * HID, BT), BT, 0, stream>>>(agg, bvec, hbuf, N);
    };

    gat_attention(a1s, a1d, b1);                               // layer 1 -> hbuf
    k_gemm<<<gemmBlocks, BT, 0, stream>>>(hbuf, w2, nullptr, xF, N);   // x2
    gat_attention(a2s, a2d, b2);                               // layer 2 -> hbuf

    // ---- pooling + classifier ----
    k_init_f32<<<cdiv(NUM_GRAPHS * HID, BT), BT, 0, stream>>>(meanacc, NUM_GRAPHS * HID, 0.f);
    k_init_u32<<<cdiv(NUM_GRAPHS * HID, BT), BT, 0, stream>>>(maxenc, NUM_GRAPHS * HID, ENC_NEG_INF);
    k_init_u32<<<1, NUM_GRAPHS, 0, stream>>>(cnt, NUM_GRAPHS, 0u);
    k_pool<<<cdiv(N * HID, BT), BT, 0, stream>>>(hbuf, batch, meanacc, maxenc, cnt, N);
    k_classifier<<<NUM_GRAPHS, 256, 0, stream>>>(meanacc, maxenc, cnt, wc1, bc1, wc2, bc2, out);
}